// PaiNNMessage_71390946394547
// MI455X (gfx1250) — compile-verified
//
#include <hip/hip_runtime.h>

#define DEV __device__ __forceinline__

constexpr int N_NODES = 20000;
constexpr int E_EDGES = 400000;
constexpr int H       = 128;
constexpr int R       = 20;
constexpr int H3      = 3 * H;
constexpr int TILE_E  = 32;      // edges per workgroup = 2 WMMA M-tiles
constexpr int LDH     = H + 4;   // +4 dword pad -> conflict-free A-frag LDS reads

typedef __attribute__((ext_vector_type(2))) float v2f;
typedef __attribute__((ext_vector_type(8))) float v8f;

DEV v8f wmma4(v2f a, v2f b, v8f c) {
  // D = A(16x4) x B(4x16) + C, full fp32
  return __builtin_amdgcn_wmma_f32_16x16x4_f32(false, a, false, b, (short)0, c,
                                               false, false);
}

DEV float fast_silu(float x) { return x * (1.0f / (1.0f + __expf(-x))); }

DEV void atomic_add_f32(float* p, float x) {
  // non-returning -> global_atomic_add_f32 without RET (STOREcnt path)
  __hip_atomic_fetch_add(p, x, __ATOMIC_RELAXED, __HIP_MEMORY_SCOPE_AGENT);
}

__global__ void zero_out_kernel(float* __restrict__ p, int n) {
  int i = blockIdx.x * blockDim.x + threadIdx.x;
  if (i < n) p[i] = 0.0f;
}

__global__ __launch_bounds__(256)
void painn_msg_kernel(const float* __restrict__ s, const float* __restrict__ v,
                      const float* __restrict__ rbf, const float* __restrict__ cut,
                      const float* __restrict__ evec,
                      const float* __restrict__ W1, const float* __restrict__ b1,
                      const float* __restrict__ W2, const float* __restrict__ b2,
                      const float* __restrict__ Wr, const float* __restrict__ br,
                      const long long* __restrict__ eidx,
                      float* __restrict__ ds, float* __restrict__ dv) {
  __shared__ float sA[TILE_E][LDH];      // gathered s[j]
  __shared__ float sHh[TILE_E][LDH];     // silu(s[j]@W1+b1)
  __shared__ float sInner[TILE_E][LDH];  // sum_d v[j][d][:]*vec[d]
  __shared__ float sRbf[TILE_E][R];
  __shared__ float sCut[TILE_E];
  __shared__ float sVec[TILE_E][3];
  __shared__ int   sI[TILE_E];
  __shared__ int   sJ[TILE_E];

  const int tid = threadIdx.x;
  const int e0  = blockIdx.x * TILE_E;

  // ---------- stage per-edge scalars ----------
  if (tid < TILE_E) {
    int  ge = e0 + tid;
    bool ok = ge < E_EDGES;
    int  gc = ok ? ge : 0;
    sI[tid]     = (int)eidx[gc];              // destination i
    sJ[tid]     = (int)eidx[E_EDGES + gc];    // source j
    sCut[tid]   = ok ? cut[gc] : 0.0f;        // cutoff=0 kills OOB edges
    sVec[tid][0] = evec[gc * 3 + 0];
    sVec[tid][1] = evec[gc * 3 + 1];
    sVec[tid][2] = evec[gc * 3 + 2];
  }
  __syncthreads();

  // ---------- gather s[j] and inner = sum_d v[j][d]*vec_d (L2 hits) ----------
  for (int k = tid; k < TILE_E * H; k += 256) {
    int e = k >> 7;        // /H
    int h = k & (H - 1);
    int j = sJ[e];
    sA[e][h] = s[j * H + h];
    sInner[e][h] = v[(j * 3 + 0) * H + h] * sVec[e][0]
                 + v[(j * 3 + 1) * H + h] * sVec[e][1]
                 + v[(j * 3 + 2) * H + h] * sVec[e][2];
  }
  for (int k = tid; k < TILE_E * R; k += 256) {
    int e = k / R, r = k - e * R;
    int ge = e0 + e;
    sRbf[e][r] = (ge < E_EDGES) ? rbf[ge * R + r] : 0.0f;
  }
  __syncthreads();

  const int wave = tid >> 5;
  const int lane = tid & 31;
  const int nl   = lane & 15;
  const int kh   = (lane < 16) ? 0 : 2;   // K sub-offset per lane half (A & B frags)
  const int mro  = (lane < 16) ? 0 : 8;   // D-matrix M offset per lane half

  // ---------- step A: h = silu(sA @ W1 + b1); wave owns cols [16w,16w+16) ----------
  {
    const int c = wave * 16 + nl;
    v8f acc0 = {}, acc1 = {};
#pragma unroll 4
    for (int k0 = 0; k0 < H; k0 += 4) {
      v2f bfr;                               // B frag shared by both M-tiles
      bfr.x = W1[(k0 + kh) * H + c];
      bfr.y = W1[(k0 + kh + 1) * H + c];
      v2f a0, a1;
      a0.x = sA[nl][k0 + kh];      a0.y = sA[nl][k0 + kh + 1];
      a1.x = sA[16 + nl][k0 + kh]; a1.y = sA[16 + nl][k0 + kh + 1];
      acc0 = wmma4(a0, bfr, acc0);
      acc1 = wmma4(a1, bfr, acc1);
    }
    float bias = b1[c];
#pragma unroll
    for (int r = 0; r < 8; ++r) {
      sHh[r + mro][c]      = fast_silu(acc0[r] + bias);
      sHh[16 + r + mro][c] = fast_silu(acc1[r] + bias);
    }
  }
  __syncthreads();

  // ---------- step B: x = (h@W2+b2) * ((rbf@Wr+br)*cutoff), fused scatter ----------
  const int c = wave * 16 + nl;   // column within each H-sized third
  v8f aS[2] = {{}, {}}, aV[2] = {{}, {}}, aU[2] = {{}, {}};
#pragma unroll 2
  for (int k0 = 0; k0 < H; k0 += 4) {
    const float* w0 = W2 + (k0 + kh) * H3;
    const float* w1 = W2 + (k0 + kh + 1) * H3;
    v2f bS, bV, bU;
    bS.x = w0[c];         bS.y = w1[c];
    bV.x = w0[H + c];     bV.y = w1[H + c];
    bU.x = w0[2 * H + c]; bU.y = w1[2 * H + c];
#pragma unroll
    for (int m = 0; m < 2; ++m) {
      v2f a;
      a.x = sHh[m * 16 + nl][k0 + kh];
      a.y = sHh[m * 16 + nl][k0 + kh + 1];
      aS[m] = wmma4(a, bS, aS[m]);
      aV[m] = wmma4(a, bV, aV[m]);
      aU[m] = wmma4(a, bU, aU[m]);
    }
  }
  // rbf @ Wr for the same three column groups (K = 20)
  v8f wS[2] = {{}, {}}, wV[2] = {{}, {}}, wU[2] = {{}, {}};
#pragma unroll
  for (int k0 = 0; k0 < R; k0 += 4) {
    const float* w0 = Wr + (k0 + kh) * H3;
    const float* w1 = Wr + (k0 + kh + 1) * H3;
    v2f bS, bV, bU;
    bS.x = w0[c];         bS.y = w1[c];
    bV.x = w0[H + c];     bV.y = w1[H + c];
    bU.x = w0[2 * H + c]; bU.y = w1[2 * H + c];
#pragma unroll
    for (int m = 0; m < 2; ++m) {
      v2f a;
      a.x = sRbf[m * 16 + nl][k0 + kh];
      a.y = sRbf[m * 16 + nl][k0 + kh + 1];
      wS[m] = wmma4(a, bS, wS[m]);
      wV[m] = wmma4(a, bV, wV[m]);
      wU[m] = wmma4(a, bU, wU[m]);
    }
  }

  const float b2S = b2[c], b2V = b2[H + c], b2U = b2[2 * H + c];
  const float brS = br[c], brV = br[H + c], brU = br[2 * H + c];
#pragma unroll
  for (int m = 0; m < 2; ++m) {
#pragma unroll
    for (int r = 0; r < 8; ++r) {
      const int   M  = m * 16 + mro + r;       // edge within tile
      const float cu = sCut[M];
      const float Ws = (wS[m][r] + brS) * cu;
      const float Wv = (wV[m][r] + brV) * cu;
      const float Wu = (wU[m][r] + brU) * cu;
      const float xss = (aS[m][r] + b2S) * Ws;
      const float xsv = (aV[m][r] + b2V) * Wv;
      const float xvv = (aU[m][r] + b2U) * Wu;
      const int   i   = sI[M];
      // merge the two dv contributions before scattering: dv += vec_d*(xsv + inner*xvv)
      const float coef = xsv + sInner[M][c] * xvv;
      atomic_add_f32(ds + i * H + c, xss);
      float* dvb = dv + i * H3;
      atomic_add_f32(dvb + 0 * H + c, sVec[M][0] * coef);
      atomic_add_f32(dvb + 1 * H + c, sVec[M][1] * coef);
      atomic_add_f32(dvb + 2 * H + c, sVec[M][2] * coef);
    }
  }
}

extern "C" void kernel_launch(void* const* d_in, const int* in_sizes, int n_in,
                              void* d_out, int out_size, void* d_ws, size_t ws_size,
                              hipStream_t stream) {
  const float*     s    = (const float*)d_in[0];
  const float*     v    = (const float*)d_in[1];
  const float*     rbf  = (const float*)d_in[2];
  const float*     cutf = (const float*)d_in[3];
  const float*     evec = (const float*)d_in[4];
  const float*     W1   = (const float*)d_in[5];
  const float*     b1   = (const float*)d_in[6];
  const float*     W2   = (const float*)d_in[7];
  const float*     b2   = (const float*)d_in[8];
  const float*     Wr   = (const float*)d_in[9];
  const float*     br   = (const float*)d_in[10];
  const long long* eidx = (const long long*)d_in[11];  // int64 [2,E]: row0=i, row1=j

  float* out = (float*)d_out;
  float* ds  = out;                   // [N,H]
  float* dv  = out + N_NODES * H;     // [N,3,H]

  // outputs are accumulated with atomics -> zero them every launch
  zero_out_kernel<<<(out_size + 255) / 256, 256, 0, stream>>>(out, out_size);

  const int nblk = (E_EDGES + TILE_E - 1) / TILE_E;   // 12500
  painn_msg_kernel<<<nblk, 256, 0, stream>>>(s, v, rbf, cutf, evec, W1, b1, W2,
                                             b2, Wr, br, eidx, ds, dv);
}